// FCGF_point_att2_ican_89575837925670
// MI455X (gfx1250) — compile-verified
//
#include <hip/hip_runtime.h>
#include <hip/hip_bf16.h>

typedef float v2f __attribute__((ext_vector_type(2)));
typedef float v8f __attribute__((ext_vector_type(8)));

#define EPS_BN 1e-5
#define EPS_NORM 1e-12f

// monotonic float <-> uint encoding for atomicMax on floats
__device__ __forceinline__ unsigned fenc(float f) {
    unsigned u = __float_as_uint(f);
    return (u >> 31) ? ~u : (u | 0x80000000u);
}
__device__ __forceinline__ float fdec(unsigned m) {
    unsigned u = (m >> 31) ? (m ^ 0x80000000u) : ~m;
    return __uint_as_float(u);
}

// ---------------- init: zero accumulators in ws ----------------
__global__ void k_init(float* z, float* pool, unsigned* menc, double* sums) {
    int t = threadIdx.x;
    if (t < 32) { z[t] = 0.0f; menc[t] = 0u; }   // enc(0)=~NaN-ish floor, < any real score enc
    for (int i = t; i < 1024; i += blockDim.x) pool[i] = 0.0f;
    if (t == 0) { sums[0] = 0.0; sums[1] = 0.0; }
}

// ---------------- pass A: WMMA mat-vec, BN sums, spill p=s*r and r ----------------
// Each wave processes 16-point tiles: C(16x16) = X(16x32) @ Bmat(32x16),
// Bmat col0 = w, col1 = 1/32, rest 0  ->  C col0 = s_raw, col1 = rowmean.
__global__ void __launch_bounds__(256) k_passA(const float* __restrict__ x,
                                               const float* __restrict__ w,
                                               const float* __restrict__ bptr,
                                               float* __restrict__ parr,
                                               float* __restrict__ rarr,
                                               double* __restrict__ sums,
                                               int N) {
    __shared__ float xch[8][32];                   // per-wave 16 s + 16 r exchange
    const int lane = threadIdx.x & 31;
    const int wv   = threadIdx.x >> 5;
    const int col  = lane & 15;                    // A row / B,C column index
    const int kb   = (lane >> 4) * 2;              // K pair base: lanes 0-15 -> K0,1 ; 16-31 -> K2,3
    const float bias = bptr[0];

    // B-matrix registers (constant across tiles): col0 = w[k], col1 = 1/32
    v2f bmat[8];
#pragma unroll
    for (int ks = 0; ks < 8; ++ks) {
        float b0 = 0.0f, b1 = 0.0f;
        if (col == 0)      { b0 = w[ks * 4 + kb]; b1 = w[ks * 4 + kb + 1]; }
        else if (col == 1) { b0 = 0.03125f;       b1 = 0.03125f; }
        v2f bm; bm.x = b0; bm.y = b1;
        bmat[ks] = bm;
    }

    const int wavesPerBlock = blockDim.x >> 5;
    long long gw      = (long long)blockIdx.x * wavesPerBlock + wv;
    long long wstride = (long long)gridDim.x * wavesPerBlock;
    const long long nTiles = (long long)(N >> 4);

    double accS = 0.0, accS2 = 0.0;

    for (long long tile = gw; tile < nTiles; tile += wstride) {
        const float* rowp = x + ((size_t)tile * 16 + (size_t)col) * 32 + kb;
        v8f c = {0.f, 0.f, 0.f, 0.f, 0.f, 0.f, 0.f, 0.f};
#pragma unroll
        for (int ks = 0; ks < 8; ++ks) {
            v2f a = *(const v2f*)(rowp + ks * 4);   // global_load_b64, A-layout per ISA 7.12.2
            c = __builtin_amdgcn_wmma_f32_16x16x4_f32(
                    false, a, false, bmat[ks], (short)0, c, false, false);
        }
        // C layout: lane holds column (lane&15); VGPR v holds row v + 8*(lane>>4)
        const int hi = (lane >> 4) * 8;
        if (col == 0) {
#pragma unroll
            for (int v = 0; v < 8; ++v) xch[wv][hi + v] = c[v];          // s rows
        } else if (col == 1) {
#pragma unroll
            for (int v = 0; v < 8; ++v) xch[wv][16 + hi + v] = c[v];     // r rows
        }
        // same-wave DS ops are ordered; compiler inserts s_wait_dscnt
        if (lane < 16) {
            float sfull = xch[wv][lane] + bias;
            float rmean = xch[wv][16 + lane];
            size_t idx = (size_t)tile * 16 + (size_t)lane;
            parr[idx] = sfull * rmean;
            rarr[idx] = rmean;
            accS  += (double)sfull;
            accS2 += (double)sfull * (double)sfull;
        }
    }

    // scalar tail (N % 16), handled by one wave
    int tail = N & 15;
    if (tail && blockIdx.x == 0 && wv == 0 && lane < tail) {
        size_t idx = ((size_t)nTiles << 4) + (size_t)lane;
        const float* rp = x + idx * 32;
        float s = bias, r = 0.0f;
        for (int k = 0; k < 32; ++k) { s = fmaf(rp[k], w[k], s); r += rp[k]; }
        r *= 0.03125f;
        parr[idx] = s * r;
        rarr[idx] = r;
        accS  += (double)s;
        accS2 += (double)s * (double)s;
    }

    // wave reduce + one f64 atomic pair per wave
    for (int off = 16; off; off >>= 1) {
        accS  += __shfl_down(accS, off);
        accS2 += __shfl_down(accS2, off);
    }
    if (lane == 0) {
        atomicAdd(&sums[0], accS);
        atomicAdd(&sums[1], accS2);
    }
}

// ---------------- stats: BN constants ----------------
__global__ void k_stats(const double* __restrict__ sums,
                        const float* __restrict__ gamma,
                        const float* __restrict__ beta,
                        float* __restrict__ consts, int N) {
    if (threadIdx.x == 0 && blockIdx.x == 0) {
        double mu  = sums[0] / (double)N;
        double var = sums[1] / (double)N - mu * mu;   // biased, as torch BN
        float c1 = gamma[0] * (float)(1.0 / sqrt(var + EPS_BN));
        float c0 = beta[0] - c1 * (float)mu;
        consts[0] = c1;
        consts[1] = c0;
    }
}

// ---------------- pass B: per-segment max of score ----------------
__global__ void __launch_bounds__(256) k_passB(const float* __restrict__ parr,
                                               const float* __restrict__ rarr,
                                               const int* __restrict__ seg,
                                               const float* __restrict__ consts,
                                               unsigned* __restrict__ menc, int N) {
    const float c1 = consts[0], c0 = consts[1];
    long long t = (long long)blockIdx.x * blockDim.x + threadIdx.x;
    long long base = t * 16;
    if (base >= N) return;
    long long end = base + 16; if (end > N) end = N;
    int cur = -1; float lmax = -3.4e38f;
    for (long long i = base; i < end; ++i) {
        int s = seg[i];
        if (s != cur) {
            if (cur >= 0) atomicMax(&menc[cur], fenc(lmax));
            cur = s; lmax = -3.4e38f;
        }
        float sc = fmaf(c1, parr[i], c0 * rarr[i]);
        lmax = fmaxf(lmax, sc);
    }
    if (cur >= 0) atomicMax(&menc[cur], fenc(lmax));
}

// ---------------- pass D: fused z = sum(e) and pooled += e*x ----------------
// group of 32 lanes = one point (lane = dim), coalesced 128B/point.
__global__ void __launch_bounds__(256) k_passD(const float* __restrict__ x,
                                               const float* __restrict__ parr,
                                               const float* __restrict__ rarr,
                                               const int* __restrict__ seg,
                                               const float* __restrict__ consts,
                                               const unsigned* __restrict__ menc,
                                               float* __restrict__ z,
                                               float* __restrict__ pool, int N) {
    const float c1 = consts[0], c0 = consts[1];
    const int lane = threadIdx.x & 31;
    const int grp  = threadIdx.x >> 5;
    const int GP = 512;                                  // points per group
    long long base = ((long long)blockIdx.x * 8 + grp) * GP;
    float acc = 0.0f, ze = 0.0f;
    int cur = -1;
    for (int t = 0; t < GP; ++t) {
        long long p = base + t;
        if (p >= N) break;
        int s = seg[p];
        if (s != cur) {
            if (cur >= 0) {
                atomicAdd(&pool[cur * 32 + lane], acc);
                if (lane == 0) atomicAdd(&z[cur], ze);
            }
            cur = s; acc = 0.0f; ze = 0.0f;
        }
        float sc = fmaf(c1, parr[p], c0 * rarr[p]);
        float e  = __expf(sc - fdec(menc[s]));
        acc = fmaf(e, x[(size_t)p * 32 + lane], acc);
        ze += e;
    }
    if (cur >= 0) {
        atomicAdd(&pool[cur * 32 + lane], acc);
        if (lane == 0) atomicAdd(&z[cur], ze);
    }
}

// ---------------- finalize: /(z*len), row L2-normalize ----------------
__global__ void k_final(const float* __restrict__ pool,
                        const float* __restrict__ z,
                        const int* __restrict__ length,
                        float* __restrict__ out) {
    int b = threadIdx.x >> 5;          // 32 waves = 32 segments
    int d = threadIdx.x & 31;          // lane = dim
    float v = pool[b * 32 + d] / (z[b] * (float)length[b]);
    float sq = v * v;
    for (int off = 16; off; off >>= 1) sq += __shfl_xor(sq, off);
    float norm = sqrtf(sq);
    out[b * 32 + d] = v / fmaxf(norm, EPS_NORM);
}

extern "C" void kernel_launch(void* const* d_in, const int* in_sizes, int n_in,
                              void* d_out, int out_size, void* d_ws, size_t ws_size,
                              hipStream_t stream) {
    const float* x      = (const float*)d_in[0];
    const float* w      = (const float*)d_in[1];
    const float* bb     = (const float*)d_in[2];
    const float* gamma  = (const float*)d_in[3];
    const float* beta   = (const float*)d_in[4];
    const int*   length = (const int*)d_in[5];
    const int*   seg    = (const int*)d_in[6];
    const int N = in_sizes[0] / 32;

    char* ws = (char*)d_ws;
    double*   sums   = (double*)(ws + 0);        // 2 f64
    float*    consts = (float*)(ws + 16);        // C1, C0
    unsigned* menc   = (unsigned*)(ws + 64);     // 32 encoded maxima
    float*    z      = (float*)(ws + 192);       // 32 softmax denominators
    float*    pool   = (float*)(ws + 320);       // 32x32 raw pooled sums
    float*    parr   = (float*)(ws + 4608);                        // N: s*r
    float*    rarr   = (float*)(ws + 4608 + (size_t)N * 4);        // N: rowmean

    k_init<<<1, 256, 0, stream>>>(z, pool, menc, sums);

    k_passA<<<1024, 256, 0, stream>>>(x, w, bb, parr, rarr, sums, N);

    k_stats<<<1, 32, 0, stream>>>(sums, gamma, beta, consts, N);

    long long nthB = ((long long)N + 15) / 16;
    int blocksB = (int)((nthB + 255) / 256);
    k_passB<<<blocksB, 256, 0, stream>>>(parr, rarr, seg, consts, menc, N);

    int blocksD = (int)(((long long)N + 4095) / 4096);
    k_passD<<<blocksD, 256, 0, stream>>>(x, parr, rarr, seg, consts, menc, z, pool, N);

    k_final<<<1, 1024, 0, stream>>>(pool, z, length, (float*)d_out);
}